// NeuralCellularAutomata_26233660244098
// MI455X (gfx1250) — compile-verified
//
#include <hip/hip_runtime.h>

typedef __attribute__((ext_vector_type(2))) float v2f;
typedef __attribute__((ext_vector_type(8))) float v8f;

#define WAVES   4
#define NTILES  (32 * 128 * 8)   // B * W * (H/16) = 32768

__device__ __forceinline__ v8f wmma4(v2f a, v2f b, v8f c) {
    return __builtin_amdgcn_wmma_f32_16x16x4_f32(false, a, false, b, (short)0, c,
                                                 false, false);
}

// GEMM-K ordering for the conv im2col: k = dy*48 + dx*16 + ci
// Patch layout: xs[r][dx][ci], row stride 52  ->  A[m][k] = xs[m*52 + dy*52 + dx*16 + ci]

__global__ __launch_bounds__(WAVES * 32) void nca_fused(
    const float* __restrict__ x,   // [32,16,128,128]
    const float* __restrict__ Wx,  // [16,16,3,3]
    const float* __restrict__ bx,  // [16]
    const float* __restrict__ Wy,  // [16,16,3,3]
    const float* __restrict__ by,  // [16]
    const float* __restrict__ W1,  // [128,48]
    const float* __restrict__ b1,  // [128]
    const float* __restrict__ W2,  // [16,128]
    const float* __restrict__ b2,  // [16]
    float* __restrict__ out)       // [32,128,128,16]  (B,W,H,O)
{
    // ---- block-shared weight staging (transposed + K-permuted to fragment order) ----
    __shared__ float s_wB[144 * 32];   // [k][o], k = dy*48+dx*16+ci ; o<16: Wx, o>=16: Wy
    __shared__ float s_B1[48 * 128];   // [k][n] = W1[n][k]
    __shared__ float s_B2[128 * 16];   // [k][n] = W2[n][k]
    __shared__ float s_bias[176];      // [0:32)=bx|by, [32:160)=b1, [160:176)=b2
    // ---- per-wave scratch ----
    __shared__ float s_xs[WAVES][18 * 52];     // patch [r][dx][ci], stride 52
    __shared__ float s_ft[WAVES][16 * 50];     // feat 16x48, row stride 50
    __shared__ float s_h [WAVES][16 * 130];    // hidden 16x128, row stride 130

    const int tid = threadIdx.x;
    const int NT  = blockDim.x;

    for (int i = tid; i < 144 * 32; i += NT) {
        int k = i >> 5, o = i & 31;
        int dy = k / 48, rem = k - dy * 48;
        int dx = rem >> 4, ci = rem & 15;
        int widx = ci * 9 + dy * 3 + dx;          // index into [16,3,3] block
        s_wB[i] = (o < 16) ? Wx[o * 144 + widx] : Wy[(o - 16) * 144 + widx];
    }
    for (int i = tid; i < 48 * 128; i += NT) {
        int k = i >> 7, n = i & 127;
        s_B1[i] = W1[n * 48 + k];
    }
    for (int i = tid; i < 128 * 16; i += NT) {
        int k = i >> 4, n = i & 15;
        s_B2[i] = W2[n * 128 + k];
    }
    if (tid < 32)  s_bias[tid]       = (tid < 16) ? bx[tid] : by[tid - 16];
    if (tid < 128) s_bias[32 + tid]  = b1[tid];
    if (tid < 16)  s_bias[160 + tid] = b2[tid];
    __syncthreads();

    const int wid  = tid >> 5;
    const int lane = tid & 31;
    const int m    = lane & 15;          // row / col index within a 16-tile
    const bool lo  = lane < 16;          // lane half selects K pair / M half
    const int kAdj = lo ? 0 : 2;
    float* xs = s_xs[wid];
    float* ft = s_ft[wid];
    float* hb = s_h[wid];
    // single loop-invariant fragment base pointers (immediate DS offsets do the rest)
    const float* aConv = xs + m * 52 + kAdj;
    const float* bConv = s_wB + kAdj * 32 + m;
    const float* aF    = ft + m * 50 + kAdj;
    const float* bF    = s_B1 + kAdj * 128 + m;
    const float* aH    = hb + m * 130 + kAdj;
    const float* bH    = s_B2 + kAdj * 16 + m;

    // Uniform trip count for every wave -> EXEC stays all-ones at every WMMA.
    for (int t = blockIdx.x * WAVES + wid; t < NTILES; t += (int)gridDim.x * WAVES) {
        const int h0 = (t & 7) << 4;          // H/16 = 8 tiles
        const int w  = (t >> 3) & 127;
        const int b  = t >> 10;

        // ---- stage patch xs[r][dx][ci] (zero-padded), 18x3x16 values ----
        for (int j = lane; j < 864; j += 32) {
            int ci = j & 15;
            int t2 = j >> 4;          // r*3 + dx
            int dx = t2 % 3;
            int r  = t2 / 3;
            int hh = h0 - 1 + r;
            int ww = w - 1 + dx;
            float v = 0.f;
            if ((unsigned)hh < 128u && (unsigned)ww < 128u)
                v = x[(((b * 16 + ci) * 128 + hh) << 7) + ww];
            xs[r * 52 + dx * 16 + ci] = v;
        }

        // ---- conv as im2col GEMM: 16x144 @ 144x32 (A shared by both N-tiles) ----
        {
            float bv0 = s_bias[m];
            float bv1 = s_bias[16 + m];
            v8f acc0 = {bv0, bv0, bv0, bv0, bv0, bv0, bv0, bv0};
            v8f acc1 = {bv1, bv1, bv1, bv1, bv1, bv1, bv1, bv1};
#pragma unroll
            for (int k4 = 0; k4 < 36; ++k4) {
                const int g  = k4 >> 2;            // (dy,dx) group
                const int dy = g / 3, dx = g - dy * 3;
                const int q  = (k4 & 3) * 4;       // ci offset within group
                const int aOff = dy * 52 + dx * 16 + q;   // compile-time immediate
                const int kL   = k4 * 4;                  // GEMM-K base
                v2f a = *(const v2f*)(aConv + aOff);      // ds_load_b64, imm offset
                v2f b0, b1f;
                b0.x  = bConv[kL * 32];
                b0.y  = bConv[kL * 32 + 32];
                b1f.x = bConv[kL * 32 + 16];
                b1f.y = bConv[kL * 32 + 48];
                acc0 = wmma4(a, b0, acc0);
                acc1 = wmma4(a, b1f, acc1);
            }
#pragma unroll
            for (int r = 0; r < 8; ++r) {
                int row = lo ? r : r + 8;
                ft[row * 50 + m]      = acc0[r];
                ft[row * 50 + 16 + m] = acc1[r];
            }
        }
        // ---- passthrough channels feat[:,32:48] = x at (h0+m, w): dx=1 column ----
#pragma unroll
        for (int ii = 0; ii < 8; ++ii) {
            int i  = ii * 32 + lane;
            int ci = i >> 4;
            int mm = i & 15;
            ft[mm * 50 + 32 + ci] = xs[(mm + 1) * 52 + 16 + ci];
        }

        // ---- MLP1: 16x48 @ 48x128 + b1, ReLU. A loaded once per K-step. ----
        {
            v8f acc[8];
#pragma unroll
            for (int n8 = 0; n8 < 8; ++n8) {
                float bv = s_bias[32 + n8 * 16 + m];
                acc[n8] = (v8f){bv, bv, bv, bv, bv, bv, bv, bv};
            }
#pragma unroll 3
            for (int k4 = 0; k4 < 12; ++k4) {
                v2f a = *(const v2f*)(aF + k4 * 4);       // ds_load_b64
#pragma unroll
                for (int n8 = 0; n8 < 8; ++n8) {
                    v2f bb;
                    bb.x = bF[k4 * 512 + n8 * 16];
                    bb.y = bF[k4 * 512 + n8 * 16 + 128];
                    acc[n8] = wmma4(a, bb, acc[n8]);
                }
            }
#pragma unroll
            for (int n8 = 0; n8 < 8; ++n8) {
#pragma unroll
                for (int r = 0; r < 8; ++r) {
                    int row = lo ? r : r + 8;
                    float vv = acc[n8][r];
                    hb[row * 130 + n8 * 16 + m] = vv > 0.f ? vv : 0.f;
                }
            }
        }

        // ---- MLP2: 16x128 @ 128x16 + b2 ----
        {
            float bv = s_bias[160 + m];
            v8f acc = {bv, bv, bv, bv, bv, bv, bv, bv};
#pragma unroll 4
            for (int k4 = 0; k4 < 32; ++k4) {
                v2f a = *(const v2f*)(aH + k4 * 4);       // ds_load_b64
                v2f bb;
                bb.x = bH[k4 * 64];
                bb.y = bH[k4 * 64 + 16];
                acc = wmma4(a, bb, acc);
            }
            // out[b][w][h0+row][col] , col = m
            float* op = out + ((((b * 128 + w) * 128 + h0)) << 4);
#pragma unroll
            for (int r = 0; r < 8; ++r) {
                int row = lo ? r : r + 8;
                op[row * 16 + m] = acc[r];
            }
        }
    }
}

extern "C" void kernel_launch(void* const* d_in, const int* in_sizes, int n_in,
                              void* d_out, int out_size, void* d_ws, size_t ws_size,
                              hipStream_t stream) {
    (void)in_sizes; (void)n_in; (void)out_size; (void)d_ws; (void)ws_size;
    const float* x  = (const float*)d_in[0];
    const float* Wx = (const float*)d_in[1];
    const float* bx = (const float*)d_in[2];
    const float* Wy = (const float*)d_in[3];
    const float* by = (const float*)d_in[4];
    const float* W1 = (const float*)d_in[5];
    const float* b1 = (const float*)d_in[6];
    const float* W2 = (const float*)d_in[7];
    const float* b2 = (const float*)d_in[8];
    float* out = (float*)d_out;

    dim3 grid(2048), block(WAVES * 32);   // 8192 waves, 4 tiles per wave
    hipLaunchKernelGGL(nca_fused, grid, block, 0, stream,
                       x, Wx, bx, Wy, by, W1, b1, W2, b2, out);
}